// FeatureEnhancementModule_68693706932308
// MI455X (gfx1250) — compile-verified
//
#include <hip/hip_runtime.h>

typedef __attribute__((ext_vector_type(2))) float v2f;
typedef __attribute__((ext_vector_type(4))) float vf4;
typedef __attribute__((ext_vector_type(8))) float v8f;

// D = A(16x4 f32) * B(4x16 f32) + C(16x16 f32)
#define WMMA_F32_4(a, b, c) \
  __builtin_amdgcn_wmma_f32_16x16x4_f32(false, (a), false, (b), (short)0, (c), false, false)

// Combined distance pair tables: 14 interaction pairs + 6 orientation pairs = 20
// CI/CJ[p] give the two landmark rows whose 16-dim distance feeds MLP input p.
__global__ __launch_bounds__(256) void fem_kernel(
    const float* __restrict__ x,   // [B,42,16]
    const float* __restrict__ W1,  // [8,20]
    const float* __restrict__ b1,  // [8]
    const float* __restrict__ W2,  // [16,8]
    const float* __restrict__ b2,  // [16]
    float* __restrict__ out)       // [B,42,16]
{
  constexpr int CI[20] = {9, 4, 8, 12, 16, 20, 0, 5, 26, 0, 21, 5, 26, 9,
                          5, 17, 17, 26, 38, 38};
  constexpr int CJ[20] = {30, 25, 29, 33, 37, 41, 21, 17, 38, 4, 25, 8, 29, 12,
                          0, 0, 5, 21, 21, 26};
  // fingertip rows {4,8,12,16,20,25,29,33,37,41}
  constexpr unsigned long long FMASK =
      (1ull << 4) | (1ull << 8) | (1ull << 12) | (1ull << 16) | (1ull << 20) |
      (1ull << 25) | (1ull << 29) | (1ull << 33) | (1ull << 37) | (1ull << 41);

  __shared__ float ldsE[8][16][16];  // per-wave 16x16 f32 tile (h, then enhanced)

  const int tid = threadIdx.x;
  const int w   = tid >> 5;   // wave in block (0..7)
  const int l   = tid & 31;   // lane
  const int m   = l & 15;     // M index (batch-in-group) / N index (feature)
  const int hh  = l >> 4;     // half-wave selector (A/B layout K split)
  const bool nvalid = (m < 8);           // layer-1 output column is real
  const float nmask = nvalid ? 1.f : 0.f;
  const int  nc = m & 7;                 // clamped (always in-bounds) W1/b1 row
  const long long wb = ((long long)blockIdx.x * 8 + w) * 16;  // wave's first batch

  // ---------------- Pass 1: 20 pairwise distances per batch element ---------
  // Lane l owns batch (wb + m). A-matrix layout for V_WMMA_F32_16X16X4_F32:
  //   VGPR0: lanes 0-15 -> K=0, lanes 16-31 -> K=2 ; VGPR1: K=1 / K=3
  // so lane computes distances p = 4t + 2*hh + {0,1}, landing directly in A[t].
  const float* xb = x + (wb + m) * 672;
  v2f A[5];
#pragma unroll
  for (int t = 0; t < 5; ++t) {
#pragma unroll
    for (int c = 0; c < 2; ++c) {
      const int p0 = 4 * t + c;                 // compile-time
      const int ip = hh ? CI[p0 + 2] : CI[p0];  // folds to 2 consts + cndmask
      const int jp = hh ? CJ[p0 + 2] : CJ[p0];
      const vf4* ri = (const vf4*)(xb + ip * 16);
      const vf4* rj = (const vf4*)(xb + jp * 16);
      float s = 0.f;
#pragma unroll
      for (int q = 0; q < 4; ++q) {
        vf4 d = ri[q] - rj[q];
        s += d.x * d.x + d.y * d.y + d.z * d.z + d.w * d.w;
      }
      const float dist = __builtin_amdgcn_sqrtf(s);  // raw v_sqrt_f32
      if (c == 0) A[t].x = dist; else A[t].y = dist;
    }
  }

  // ---------------- Layer 1: [16x20] @ W1^T[20x8] + b1, via 5 chained WMMAs -
  // C/D layout: lane l, VGPR v -> M = v + 8*(l/16), N = l%16. C preloaded with b1[N].
  // Weight loads use clamped indices (always in-bounds) + multiply-mask so the
  // compiler emits plain unconditional global_load + v_cndmask, no EXEC branches.
  v8f C1;
  const float bias1 = b1[nc] * nmask;
#pragma unroll
  for (int v = 0; v < 8; ++v) C1[v] = bias1;
#pragma unroll
  for (int t = 0; t < 5; ++t) {
    const int k0 = 4 * t + 2 * hh;  // B layout mirrors A: VGPR0 holds K=k0 row
    v2f Bt;
    Bt.x = W1[nc * 20 + k0] * nmask;      // W1^T[k][n] = W1[n*20+k]
    Bt.y = W1[nc * 20 + k0 + 1] * nmask;
    C1 = WMMA_F32_4(A[t], Bt, C1);
  }
  // ReLU on accumulators (columns n>=8 are exactly 0: zero weights + zero bias)
#pragma unroll
  for (int v = 0; v < 8; ++v) C1[v] = fmaxf(C1[v], 0.f);

  // Re-layout h from C/D layout to A layout through this wave's LDS tile.
#pragma unroll
  for (int v = 0; v < 8; ++v) ldsE[w][v + 8 * hh][m] = C1[v];  // [batch][feat]
  __syncthreads();

  v2f A2[2];
#pragma unroll
  for (int t = 0; t < 2; ++t) {
    const int k0 = 4 * t + 2 * hh;
    A2[t].x = ldsE[w][m][k0];      // A2[M=batch m][K=hidden k]
    A2[t].y = ldsE[w][m][k0 + 1];
  }
  __syncthreads();

  // ---------------- Layer 2: [16x8] @ W2^T[8x16] + b2, via 2 WMMAs ----------
  v8f E;
  const float bias2 = b2[m];       // m < 16: always in-bounds
#pragma unroll
  for (int v = 0; v < 8; ++v) E[v] = bias2;
#pragma unroll
  for (int t = 0; t < 2; ++t) {
    const int k0 = 4 * t + 2 * hh;
    v2f Bt;
    Bt.x = W2[m * 8 + k0];         // W2^T[k][n] = W2[n*8+k]; max idx 127 < 128
    Bt.y = W2[m * 8 + k0 + 1];
    E = WMMA_F32_4(A2[t], Bt, E);
  }

  // enhanced -> LDS in [batch][feature] layout for the copy pass
#pragma unroll
  for (int v = 0; v < 8; ++v) ldsE[w][v + 8 * hh][m] = E[v];
  __syncthreads();

  // ---------------- Pass 2: streamed copy + fingertip scatter-add -----------
  // Wave's chunk is contiguous: 16 batches * 168 float4 = 2688 float4.
  const vf4* xin4 = (const vf4*)(x + wb * 672);
  vf4* out4 = (vf4*)(out + wb * 672);
#pragma unroll 4
  for (int it = 0; it < 84; ++it) {
    const int local = it * 32 + l;       // coalesced: 512B per wave per iter
    const int bl  = local / 168;         // batch within wave group
    const int rem = local - bl * 168;
    const int row = rem >> 2;            // landmark row (0..41)
    const int q   = rem & 3;             // which float4 of the 16-float row
    vf4 v = xin4[local];
    if ((FMASK >> row) & 1ull) {
      const vf4 e = *(const vf4*)&ldsE[w][bl][q * 4];
      v += e;
    }
    out4[local] = v;
  }
}

extern "C" void kernel_launch(void* const* d_in, const int* in_sizes, int n_in,
                              void* d_out, int out_size, void* d_ws, size_t ws_size,
                              hipStream_t stream) {
  const float* x  = (const float*)d_in[0];
  const float* W1 = (const float*)d_in[1];
  const float* b1 = (const float*)d_in[2];
  const float* W2 = (const float*)d_in[3];
  const float* b2 = (const float*)d_in[4];
  float* out = (float*)d_out;

  const int Btot = in_sizes[0] / (42 * 16);  // 131072
  const int blocks = Btot / 128;             // 128 batch elements per block
  fem_kernel<<<blocks, 256, 0, stream>>>(x, W1, b1, W2, b2, out);
}